// MMDLoss_60550448939558
// MI455X (gfx1250) — compile-verified
//
#include <hip/hip_runtime.h>

// MMD loss with multi-bandwidth Gaussian kernels for MI455X (gfx1250, wave32).
// GEMM core: V_WMMA_F32_16X16X4_F32, 4-way A-fragment reuse (16x64 strip per
// wave) + explicit double-buffered fragment loads so WMMA issue overlaps
// memory latency instead of serializing on s_wait_loadcnt 0.

typedef float v2f __attribute__((ext_vector_type(2)));
typedef float v8f __attribute__((ext_vector_type(8)));

#define NPTS 4096
#define DIMS 256
#define NPAIR 16773120.0   // n*n - n, n = 4096
#define EPS   1e-5f

// Workspace layout:
//   [0      , 16384): xx row norms (4096 f32)
//   [16384  , 32768): yy row norms (4096 f32)
//   [32768  , 32816): 6 doubles: l2sum[3] (XX,YY,XY), ksum[3] (XX,YY,XY)

__global__ void mmd_init_accum(double* __restrict__ acc) {
    if (threadIdx.x < 6) acc[threadIdx.x] = 0.0;
}

__global__ void mmd_row_norms(const float* __restrict__ X,
                              const float* __restrict__ Y,
                              float* __restrict__ norms) {
    __shared__ float s[256];
    const float* M = blockIdx.y ? Y : X;
    const int row = blockIdx.x;
    float v = M[row * DIMS + threadIdx.x];
    s[threadIdx.x] = v * v;
    __syncthreads();
    for (int off = 128; off > 0; off >>= 1) {
        if ((int)threadIdx.x < off) s[threadIdx.x] += s[threadIdx.x + off];
        __syncthreads();
    }
    if (threadIdx.x == 0) norms[blockIdx.y * NPTS + row] = s[0];
}

#define WMMA_F32(a, b, c) \
    __builtin_amdgcn_wmma_f32_16x16x4_f32(false, (a), false, (b), (short)0, (c), false, false)

// One wave computes a 16x64 strip: A[m0:m0+16,:] * B[n0:n0+64,:]^T as four
// 16x16 tiles sharing one A fragment per K-step, with 2-deep fragment
// pipelining (loads for step k+4 issue before WMMAs for step k).
// Fragment layout (V_WMMA_F32_16X16X4_F32, wave32):
//   frag[v] = Row[base + (lane&15)][k + 2*(lane>>4) + v]   (aligned float2)
__device__ __forceinline__ void wmma_dot_strip(const float* __restrict__ A,
                                               const float* __restrict__ B,
                                               int m0, int n0, int lane,
                                               v8f acc[4]) {
    const int half = lane >> 4;
    const int l    = lane & 15;
    const float* arow = A + (size_t)(m0 + l) * DIMS + 2 * half;
    const float* b0   = B + (size_t)(n0 +  0 + l) * DIMS + 2 * half;
    const float* b1   = B + (size_t)(n0 + 16 + l) * DIMS + 2 * half;
    const float* b2   = B + (size_t)(n0 + 32 + l) * DIMS + 2 * half;
    const float* b3   = B + (size_t)(n0 + 48 + l) * DIMS + 2 * half;

#pragma unroll
    for (int j = 0; j < 4; ++j)
        acc[j] = (v8f){0.f, 0.f, 0.f, 0.f, 0.f, 0.f, 0.f, 0.f};

    // Prologue: stage K-step 0.
    v2f a  = *(const v2f*)(arow);
    v2f f0 = *(const v2f*)(b0);
    v2f f1 = *(const v2f*)(b1);
    v2f f2 = *(const v2f*)(b2);
    v2f f3 = *(const v2f*)(b3);

#pragma unroll 8
    for (int k = 4; k < DIMS; k += 4) {
        // Issue next K-step's loads before consuming the current fragments.
        v2f an_ = *(const v2f*)(arow + k);
        v2f g0  = *(const v2f*)(b0 + k);
        v2f g1  = *(const v2f*)(b1 + k);
        v2f g2  = *(const v2f*)(b2 + k);
        v2f g3  = *(const v2f*)(b3 + k);

        acc[0] = WMMA_F32(a, f0, acc[0]);
        acc[1] = WMMA_F32(a, f1, acc[1]);
        acc[2] = WMMA_F32(a, f2, acc[2]);
        acc[3] = WMMA_F32(a, f3, acc[3]);

        a = an_; f0 = g0; f1 = g1; f2 = g2; f3 = g3;
    }

    // Epilogue: final K-step.
    acc[0] = WMMA_F32(a, f0, acc[0]);
    acc[1] = WMMA_F32(a, f1, acc[1]);
    acc[2] = WMMA_F32(a, f2, acc[2]);
    acc[3] = WMMA_F32(a, f3, acc[3]);
}

__device__ __forceinline__ float wave_reduce_add(float v) {
#pragma unroll
    for (int off = 16; off > 0; off >>= 1)
        v += __shfl_xor(v, off, 32);
    return v;
}

// Pass 1: accumulate sum(max(l2,0)) per matrix (z = 0:XX, 1:YY, 2:XY).
__global__ void mmd_pass1(const float* __restrict__ X,
                          const float* __restrict__ Y,
                          const float* __restrict__ norms,
                          double* __restrict__ l2sum) {
    const int z = blockIdx.z;
    const float* A  = (z == 1) ? Y : X;
    const float* B  = (z == 0) ? X : Y;
    const float* an = norms + ((z == 1) ? NPTS : 0);
    const float* bn = norms + ((z == 0) ? 0 : NPTS);

    const int lane = threadIdx.x & 31;
    const int wave = threadIdx.x >> 5;
    const int m0 = blockIdx.x * 64  + (wave & 3) * 16;
    const int n0 = blockIdx.y * 128 + (wave >> 2) * 64;

    v8f acc[4];
    wmma_dot_strip(A, B, m0, n0, lane, acc);

    const int half = lane >> 4, l = lane & 15;
    float anv[8];
#pragma unroll
    for (int r = 0; r < 8; ++r) anv[r] = an[m0 + r + 8 * half];

    float partial = 0.f;
#pragma unroll
    for (int j = 0; j < 4; ++j) {
        const float bnv = bn[n0 + 16 * j + l];
#pragma unroll
        for (int r = 0; r < 8; ++r) {
            float l2 = anv[r] + bnv - 2.0f * acc[j][r];
            partial += fmaxf(l2, 0.0f);
        }
    }
    partial = wave_reduce_add(partial);
    if (lane == 0) atomicAdd(&l2sum[z], (double)partial);
}

// Pass 2: recompute l2 tiles, accumulate sum over the 5-bandwidth kernel.
__global__ void mmd_pass2(const float* __restrict__ X,
                          const float* __restrict__ Y,
                          const float* __restrict__ norms,
                          const double* __restrict__ l2sum,
                          double* __restrict__ ksum) {
    const int z = blockIdx.z;
    const float* A  = (z == 1) ? Y : X;
    const float* B  = (z == 0) ? X : Y;
    const float* an = norms + ((z == 1) ? NPTS : 0);
    const float* bn = norms + ((z == 0) ? 0 : NPTS);

    // sigma = sum(l2)/(n^2-n) / kernel_mul^(kernel_num//2); bandwidths sigma*8^i.
    const float sigma = (float)(l2sum[z] / NPAIR / 64.0);
    float c[5];
    float bw = sigma;
#pragma unroll
    for (int i = 0; i < 5; ++i) {
        c[i] = 1.0f / (bw + EPS);
        bw *= 8.0f;
    }

    const int lane = threadIdx.x & 31;
    const int wave = threadIdx.x >> 5;
    const int m0 = blockIdx.x * 64  + (wave & 3) * 16;
    const int n0 = blockIdx.y * 128 + (wave >> 2) * 64;

    v8f acc[4];
    wmma_dot_strip(A, B, m0, n0, lane, acc);

    const int half = lane >> 4, l = lane & 15;
    float anv[8];
#pragma unroll
    for (int r = 0; r < 8; ++r) anv[r] = an[m0 + r + 8 * half];

    float partial = 0.f;
#pragma unroll
    for (int j = 0; j < 4; ++j) {
        const float bnv = bn[n0 + 16 * j + l];
#pragma unroll
        for (int r = 0; r < 8; ++r) {
            float l2 = fmaxf(anv[r] + bnv - 2.0f * acc[j][r], 0.0f);
#pragma unroll
            for (int i = 0; i < 5; ++i)
                partial += __expf(-l2 * c[i]);
        }
    }
    partial = wave_reduce_add(partial);
    if (lane == 0) atomicAdd(&ksum[z], (double)partial);
}

__global__ void mmd_finalize(const double* __restrict__ ksum,
                             float* __restrict__ out) {
    // mean(Kxx) + mean(Kyy) - mean(Kxy) - mean(Kyx); Kyx mean == Kxy mean.
    const double inv = 1.0 / ((double)NPTS * (double)NPTS);
    double loss = (ksum[0] + ksum[1] - 2.0 * ksum[2]) * inv;
    out[0] = (float)(loss > 0.0 ? loss : 0.0);
}

extern "C" void kernel_launch(void* const* d_in, const int* in_sizes, int n_in,
                              void* d_out, int out_size, void* d_ws, size_t ws_size,
                              hipStream_t stream) {
    const float* X = (const float*)d_in[0];
    const float* Y = (const float*)d_in[1];
    float* out = (float*)d_out;

    float*  norms = (float*)d_ws;                           // 8192 floats
    double* accum = (double*)((char*)d_ws + 32768);         // 6 doubles
    double* l2sum = accum;                                  // [0..2]
    double* ksum  = accum + 3;                              // [3..5]

    mmd_init_accum<<<1, 64, 0, stream>>>(accum);
    mmd_row_norms<<<dim3(NPTS, 2), 256, 0, stream>>>(X, Y, norms);

    dim3 grid(NPTS / 64, NPTS / 128, 3);  // 64 x 32 x 3 blocks, 8 waves each
    mmd_pass1<<<grid, 256, 0, stream>>>(X, Y, norms, l2sum);
    mmd_pass2<<<grid, 256, 0, stream>>>(X, Y, norms, l2sum, ksum);

    mmd_finalize<<<1, 1, 0, stream>>>(ksum, out);
}